// Model_6579889897868
// MI455X (gfx1250) — compile-verified
//
#include <hip/hip_runtime.h>
#include <hip/hip_bf16.h>
#include <math.h>

// ---------------- model dims ----------------
#define BB       32
#define LL       512
#define CC       21
#define D_MODEL  256
#define D_STATE  16
#define D_CONV   4
#define D_INNER  512
#define DT_RANK  16
#define PATCH    16
#define STRIDE_  8
#define PADP     8
#define PP       65                 // patches
#define PRED     96
#define NSEQ     (BB*CC)            // 672
#define NROW     (NSEQ*PP)          // 43680
#define HEAD_NF  (D_MODEL*PP)       // 16640

typedef __attribute__((ext_vector_type(2))) float v2f;
typedef __attribute__((ext_vector_type(8))) float v8f;
typedef __attribute__((ext_vector_type(4))) int   v4i;

// ---- gfx1250 async copy to LDS (guarded: degrades to sync staging if absent) ----
#if defined(__has_builtin)
#if __has_builtin(__builtin_amdgcn_global_load_async_to_lds_b128)
#define HAVE_ASYNC_LDS 1
#endif
#endif
#ifndef HAVE_ASYNC_LDS
#define HAVE_ASYNC_LDS 0
#endif

__device__ __forceinline__ void async_cp16(const float* gp, float* lp) {
#if HAVE_ASYNC_LDS
  __builtin_amdgcn_global_load_async_to_lds_b128(
      (__attribute__((address_space(1))) v4i*)gp,
      (__attribute__((address_space(3))) v4i*)lp, 0, 0);
#else
  *(float4*)lp = *(const float4*)gp;
#endif
}

__device__ __forceinline__ void async_wait() {
#if HAVE_ASYNC_LDS
#if __has_builtin(__builtin_amdgcn_s_wait_asynccnt)
  __builtin_amdgcn_s_wait_asynccnt(0);
#else
  asm volatile("s_wait_asynccnt 0x0" ::: "memory");
#endif
#endif
}

// ---------------- 1) RevIN stats ----------------
__global__ void revin_stats(const float* __restrict__ x, float* __restrict__ meanv,
                            float* __restrict__ stdv) {
  const int n = blockIdx.x;              // n = b*C + c
  const int b = n / CC, c = n % CC;
  __shared__ float s1[256], s2[256];
  float a = 0.f, q = 0.f;
  for (int l = threadIdx.x; l < LL; l += 256) {
    float v = x[((size_t)b*LL + l)*CC + c];
    a += v; q += v*v;
  }
  s1[threadIdx.x] = a; s2[threadIdx.x] = q;
  __syncthreads();
  for (int st = 128; st > 0; st >>= 1) {
    if ((int)threadIdx.x < st) {
      s1[threadIdx.x] += s1[threadIdx.x + st];
      s2[threadIdx.x] += s2[threadIdx.x + st];
    }
    __syncthreads();
  }
  if (threadIdx.x == 0) {
    float m  = s1[0] * (1.f/(float)LL);
    float var = s2[0] * (1.f/(float)LL) - m*m;
    meanv[n] = m;
    stdv[n]  = sqrtf(var + 1e-5f);
  }
}

// ---------------- 2) patchify: xp[n,p,d] ----------------
__global__ void patchify(const float* __restrict__ x, const float* __restrict__ meanv,
                         const float* __restrict__ stdv, const float* __restrict__ Wp,
                         const float* __restrict__ bp, float* __restrict__ xp) {
  const int n = blockIdx.x;              // 0..671
  const int p = blockIdx.y;              // 0..64
  const int b = n / CC, c = n % CC;
  __shared__ float win[PATCH];
  if (threadIdx.x < PATCH) {
    int l = p*STRIDE_ - PADP + (int)threadIdx.x;
    float v = 0.f;
    if (l >= 0 && l < LL)
      v = (x[((size_t)b*LL + l)*CC + c] - meanv[n]) / stdv[n];
    win[threadIdx.x] = v;
  }
  __syncthreads();
  const int d = threadIdx.x;             // 256 threads
  float acc = bp[d];
  #pragma unroll
  for (int k = 0; k < PATCH; ++k) acc += Wp[d*PATCH + k] * win[k];
  xp[((size_t)(n*PP + p))*D_MODEL + d] = acc;
}

// ---------------- fp32 WMMA GEMM with LDS-staged panels ----------------
// C[M,N] = A[M,K] @ B[K,N] (row-major). Block = 8 waves; block tile = 128(M) x 16(N);
// each wave owns one 16x16 tile. A(128xKC) and B(KCx16) are staged to LDS each
// K-chunk with gfx1250 async-to-LDS b128 copies (coalesced), then 8 waves consume
// the shared B panel. EPI==1: C = softplus(C + bias[col]).
template<int KC, int EPI>
__global__ void wmma_gemm_f32_lds(const float* __restrict__ A, const float* __restrict__ B,
                                  float* __restrict__ C, const float* __restrict__ bias,
                                  int M, int N, int K, int lda, int ldb, int ldc) {
  __shared__ float lA[128][KC];    // row stride KC floats -> 16B-aligned float4 slots
  __shared__ float lB[KC][16];

  const int wave   = threadIdx.x >> 5;
  const int lane   = threadIdx.x & 31;
  const int tilesM = M >> 4;
  const int tn     = blockIdx.x << 4;
  const int tileM  = blockIdx.y * 8 + wave;
  const bool active = (tileM < tilesM);
  const int tmc    = (active ? tileM : (tilesM - 1)) << 4;   // clamped tile row base
  const int rowBase = blockIdx.y * 128;

  const int lrow  = (wave << 4) + (lane & 15);   // this wave's local A row
  const int colc  = tn + (lane & 15);            // output / B column
  const int khalf = (lane >> 4) << 1;            // 0 or 2

  v8f acc = {};
  for (int kk = 0; kk < K; kk += KC) {
    __syncthreads();
    // ---- stage A: 128 x KC (float4 per thread-slot, coalesced rows) ----
    {
      const int nA = (128 * KC) / 4;
      for (int t = threadIdx.x; t < nA; t += 256) {
        int lin = t * 4;
        int r   = lin / KC;
        int ck  = lin % KC;
        int gr  = rowBase + r; if (gr > M - 1) gr = M - 1;
        async_cp16(A + (size_t)gr * lda + kk + ck, &lA[r][ck]);
      }
    }
    // ---- stage B: KC x 16 ----
    {
      const int nB = (KC * 16) / 4;              // 64 (KC=16) or 128 (KC=32) slots
      if ((int)threadIdx.x < nB) {
        int r  = threadIdx.x >> 2;
        int cg = (threadIdx.x & 3) << 2;
        async_cp16(B + (size_t)(kk + r) * ldb + tn + cg, &lB[r][cg]);
      }
    }
    async_wait();
    __syncthreads();

    // ---- 16x16x4 WMMA over the chunk (KC/4 back-to-back matrix ops) ----
    #pragma unroll
    for (int k4 = 0; k4 < KC; k4 += 4) {
      const int kb = k4 + khalf;
      v2f a, b;
      a.x = lA[lrow][kb];
      a.y = lA[lrow][kb + 1];
      b.x = lB[kb][lane & 15];
      b.y = lB[kb + 1][lane & 15];
      acc = __builtin_amdgcn_wmma_f32_16x16x4_f32(
          false, a, false, b, (short)0, acc, false, false);
    }
  }

  if (active) {
    const int rbase = tmc + ((lane >> 4) << 3);
    #pragma unroll
    for (int i = 0; i < 8; ++i) {
      float v = acc[i];
      if (EPI == 1) {
        v += bias[colc];
        v = (v > 20.f) ? v : logf(1.f + __expf(v));   // softplus
      }
      C[(size_t)(rbase + i) * ldc + colc] = v;
    }
  }
}

// ---------------- 4) causal depthwise conv + SiLU ----------------
__global__ void dwconv_silu(const float* __restrict__ xz, const float* __restrict__ cw,
                            const float* __restrict__ cb, float* __restrict__ xmc) {
  const int n  = blockIdx.x;
  const int p  = blockIdx.y;
  const int ch = threadIdx.x;            // 512 threads
  float acc = cb[ch];
  #pragma unroll
  for (int j = 0; j < D_CONV; ++j) {
    int pp = p - (D_CONV - 1) + j;
    if (pp >= 0)
      acc += cw[ch*D_CONV + j] * xz[((size_t)(n*PP + pp))*(2*D_INNER) + ch];
  }
  float s = acc / (1.f + __expf(-acc));
  xmc[((size_t)(n*PP + p))*D_INNER + ch] = s;
}

// ---------------- 7) selective scan + gate ----------------
__global__ void selective_scan(const float* __restrict__ xmc, const float* __restrict__ delta,
                               const float* __restrict__ dbc, const float* __restrict__ xz,
                               const float* __restrict__ A_log, const float* __restrict__ Dv,
                               float* __restrict__ ybuf) {
  const int n = blockIdx.x;
  const int d = threadIdx.x;             // 512 threads
  __shared__ float sB[D_STATE], sC[D_STATE];
  float h[D_STATE], Arow[D_STATE];
  #pragma unroll
  for (int s = 0; s < D_STATE; ++s) {
    h[s] = 0.f;
    Arow[s] = -__expf(A_log[d*D_STATE + s]);
  }
  const float Dp = Dv[d];
  for (int p = 0; p < PP; ++p) {
    const int r = n*PP + p;
    __syncthreads();
    if (threadIdx.x < D_STATE)
      sB[threadIdx.x] = dbc[(size_t)r*48 + DT_RANK + threadIdx.x];
    else if (threadIdx.x < 2*D_STATE)
      sC[threadIdx.x - D_STATE] = dbc[(size_t)r*48 + DT_RANK + D_STATE + (threadIdx.x - D_STATE)];
    __syncthreads();
    const float u  = xmc[(size_t)r*D_INNER + d];
    const float dl = delta[(size_t)r*D_INNER + d];
    const float du = dl * u;
    float y = 0.f;
    #pragma unroll
    for (int s = 0; s < D_STATE; ++s) {
      h[s] = __expf(dl * Arow[s]) * h[s] + du * sB[s];
      y += h[s] * sC[s];
    }
    y += u * Dp;
    const float zv = xz[(size_t)r*(2*D_INNER) + D_INNER + d];
    const float sz = zv / (1.f + __expf(-zv));   // silu gate
    ybuf[(size_t)r*D_INNER + d] = y * sz;
  }
}

// ---------------- 10) de-normalize + transpose to (B, PRED, C) ----------------
__global__ void head_denorm(const float* __restrict__ headb, const float* __restrict__ meanv,
                            const float* __restrict__ stdv, float* __restrict__ out) {
  const int idx = blockIdx.x * blockDim.x + threadIdx.x;
  if (idx >= BB*PRED*CC) return;
  const int c = idx % CC;
  const int t = (idx / CC) % PRED;
  const int b = idx / (CC*PRED);
  const int n = b*CC + c;
  out[idx] = headb[(size_t)n*PRED + t] * stdv[n] + meanv[n];
}

// ---------------- launcher ----------------
static inline void launch_gemm(const float* A, const float* B, float* C, const float* bias,
                               int M, int N, int K, int lda, int ldb, int ldc, int epi,
                               hipStream_t stream) {
  dim3 grid(N >> 4, ((M >> 4) + 7) / 8);
  if (K % 32 == 0) {
    if (epi) wmma_gemm_f32_lds<32,1><<<grid, 256, 0, stream>>>(A,B,C,bias,M,N,K,lda,ldb,ldc);
    else     wmma_gemm_f32_lds<32,0><<<grid, 256, 0, stream>>>(A,B,C,bias,M,N,K,lda,ldb,ldc);
  } else {
    if (epi) wmma_gemm_f32_lds<16,1><<<grid, 256, 0, stream>>>(A,B,C,bias,M,N,K,lda,ldb,ldc);
    else     wmma_gemm_f32_lds<16,0><<<grid, 256, 0, stream>>>(A,B,C,bias,M,N,K,lda,ldb,ldc);
  }
}

extern "C" void kernel_launch(void* const* d_in, const int* in_sizes, int n_in,
                              void* d_out, int out_size, void* d_ws, size_t ws_size,
                              hipStream_t stream) {
  const float* x_enc  = (const float*)d_in[0];
  const float* Wp     = (const float*)d_in[4];
  const float* bp     = (const float*)d_in[5];
  const float* W_in   = (const float*)d_in[6];
  const float* conv_w = (const float*)d_in[7];
  const float* conv_b = (const float*)d_in[8];
  const float* W_xprj = (const float*)d_in[9];
  const float* W_dt   = (const float*)d_in[10];
  const float* b_dt   = (const float*)d_in[11];
  const float* A_log  = (const float*)d_in[12];
  const float* Dvec   = (const float*)d_in[13];
  const float* W_out  = (const float*)d_in[14];
  const float* W_head = (const float*)d_in[15];
  float* out = (float*)d_out;

  // workspace layout (floats)
  float* ws = (float*)d_ws;
  size_t o = 0;
  float* meanv = ws + o; o += NSEQ;
  float* stdv  = ws + o; o += NSEQ;
  o = (o + 255) & ~(size_t)255;
  float* xp    = ws + o; o += (size_t)NROW * D_MODEL;        // reused as `outb`
  float* xz    = ws + o; o += (size_t)NROW * 2 * D_INNER;
  float* xmc   = ws + o; o += (size_t)NROW * D_INNER;
  float* dbc   = ws + o; o += (size_t)NROW * 48;
  float* delta = ws + o; o += (size_t)NROW * D_INNER;
  float* ybuf  = ws + o; o += (size_t)NROW * D_INNER;
  float* headb = ws + o; o += (size_t)NSEQ * PRED;
  float* outb  = xp;   // xp dead after GEMM1

  // 1) RevIN statistics
  revin_stats<<<NSEQ, 256, 0, stream>>>(x_enc, meanv, stdv);
  // 2) patchify
  patchify<<<dim3(NSEQ, PP), 256, 0, stream>>>(x_enc, meanv, stdv, Wp, bp, xp);
  // 3) xz = xp @ W_in                      (43680 x 1024 x 256)
  launch_gemm(xp, W_in, xz, nullptr, NROW, 2*D_INNER, D_MODEL,
              D_MODEL, 2*D_INNER, 2*D_INNER, 0, stream);
  // 4) causal depthwise conv + SiLU
  dwconv_silu<<<dim3(NSEQ, PP), D_INNER, 0, stream>>>(xz, conv_w, conv_b, xmc);
  // 5) dbc = xmc @ W_xproj                 (43680 x 48 x 512)
  launch_gemm(xmc, W_xprj, dbc, nullptr, NROW, 48, D_INNER,
              D_INNER, 48, 48, 0, stream);
  // 6) delta = softplus(dt @ W_dt + b_dt)  (43680 x 512 x 16)
  launch_gemm(dbc, W_dt, delta, b_dt, NROW, D_INNER, DT_RANK,
              48, D_INNER, D_INNER, 1, stream);
  // 7) selective scan + z-gate
  selective_scan<<<NSEQ, D_INNER, 0, stream>>>(xmc, delta, dbc, xz, A_log, Dvec, ybuf);
  // 8) outb = ybuf @ W_out                 (43680 x 256 x 512)
  launch_gemm(ybuf, W_out, outb, nullptr, NROW, D_MODEL, D_INNER,
              D_INNER, D_MODEL, D_MODEL, 0, stream);
  // 9) head: (672 x 96 x 16640); outb rows (n*65+p) are already the flattened layout
  launch_gemm(outb, W_head, headb, nullptr, NSEQ, PRED, HEAD_NF,
              HEAD_NF, PRED, PRED, 0, stream);
  // 10) de-normalize + transpose
  head_denorm<<<(BB*PRED*CC + 255)/256, 256, 0, stream>>>(headb, meanv, stdv, out);
}